// Finer_36051955483031
// MI455X (gfx1250) — compile-verified
//
#include <hip/hip_runtime.h>
#include <hip/hip_bf16.h>

// Problem constants (match reference setup_inputs)
constexpr int kB  = 4;
constexpr int kNB = 2048;
constexpr int kBS = 32;
constexpr int kD  = 256;
constexpr int kNF = 512;
constexpr int kNC = 1024;

constexpr int kD4 = kD / 4;                 // 64 float4 per row
constexpr int kTile4 = kBS * kD4;           // 2048 float4 per fine tile
constexpr int kThreads = 256;               // 8 waves (wave32)

typedef float f4_t __attribute__((ext_vector_type(4)));

#if __has_builtin(__builtin_amdgcn_global_load_async_to_lds_b128)
#define FINER_HAVE_ASYNC_LDS 1
typedef int v4i_t __attribute__((ext_vector_type(4)));
typedef __attribute__((address_space(1))) v4i_t glob_v4i_t;   // global
typedef __attribute__((address_space(3))) v4i_t lds_v4i_t;    // LDS
#else
#define FINER_HAVE_ASYNC_LDS 0
#endif

__device__ __forceinline__ void finer_wait_asynccnt0() {
#if __has_builtin(__builtin_amdgcn_s_wait_asynccnt)
    __builtin_amdgcn_s_wait_asynccnt(0);
#else
    asm volatile("s_wait_asynccnt 0" ::: "memory");
#endif
}

// One workgroup per (b, f) fine block. Resolves the double indirection once
// (scalarized by the compiler since it is block-uniform), stages the 1KB
// broadcast coarse row into LDS via global_load_async_to_lds_b128, then
// streams the 32KB bank tile: NT b128 load -> v_sub -> NT b128 store.
__global__ __launch_bounds__(kThreads) void finer_fine_kernel(
    const float* __restrict__ fine_token_mask,      // [B, NB*BS]
    const float* __restrict__ coarse_token_states,  // [B, NB, D]
    const int*   __restrict__ fine_block_indices,   // [B, NF]
    const float* __restrict__ fine_block_scores,    // [B, NF]
    const int*   __restrict__ cache_indice_table,   // [B, NB]
    const float* __restrict__ cache_bank,           // [B, NB, BS, D]
    float* __restrict__ o_fs,                       // [B, NF*BS, D]
    float* __restrict__ o_fm,                       // [B, NF*BS]
    float* __restrict__ o_fsc,                      // [B, NF*BS]
    float* __restrict__ o_fd)                       // [B, NF*BS]
{
    __shared__ f4_t s_row[kD4];                     // coarse row for this block

    const int tid = threadIdx.x;
    const int blk = blockIdx.x;                     // 0 .. B*NF-1
    const int b   = blk / kNF;
    const int f   = blk - b * kNF;

    const int j = fine_block_indices[(size_t)b * kNF + f];     // block id
    const int k = cache_indice_table[(size_t)b * kNB + j];     // bank slot

    const f4_t* __restrict__ src4 =
        (const f4_t*)(coarse_token_states + ((size_t)b * kNB + j) * kD);
    const f4_t* __restrict__ bank4 =
        (const f4_t*)(cache_bank + ((size_t)b * kNB + k) * (size_t)(kBS * kD));
    f4_t* __restrict__ dst4 =
        (f4_t*)(o_fs + ((size_t)b * kNF + f) * (size_t)(kBS * kD));

    // Cover the whole 32KB tile with WGP-scope prefetches: 256 threads x 128B.
    __builtin_prefetch((const void*)(bank4 + (tid << 3)), 0, 3);

    // Stage the 1KB coarse row into LDS. Waves 0-1 (tid 0..63) have EXEC
    // all-1s, each lane moves 16B via the async engine (ASYNCcnt-tracked).
    if (tid < kD4) {
#if FINER_HAVE_ASYNC_LDS
        __builtin_amdgcn_global_load_async_to_lds_b128(
            (glob_v4i_t*)(src4 + tid),
            (lds_v4i_t*)(&s_row[tid]),
            /*offset=*/0, /*cpol=*/0);
        finer_wait_asynccnt0();      // issuing waves ensure LDS writes landed
#else
        s_row[tid] = src4[tid];
#endif
    }
    __syncthreads();

    // col4 = (tid + 256*i) & 63 is loop-invariant: one LDS broadcast read.
    const f4_t c = s_row[tid & (kD4 - 1)];

    // 32KB tile: 2048 float4, 256 threads -> 8 per thread, fully coalesced.
    // Read-once / write-once streams -> nontemporal (TH_NT) to spare WGP$.
#pragma unroll
    for (int i = 0; i < kTile4 / kThreads; ++i) {
        const int t  = tid + i * kThreads;     // 0..2047
        const f4_t v = __builtin_nontemporal_load(bank4 + t);
        __builtin_nontemporal_store(c - v, dst4 + t);
    }

    // Per-token mask / score / dups for this fine block (32 tokens).
    if (tid < kBS) {
        const size_t o = ((size_t)b * kNF + f) * kBS + tid;
        o_fm[o]  = fine_token_mask[((size_t)b * kNB + j) * kBS + tid];
        o_fsc[o] = fine_block_scores[(size_t)b * kNF + f];
        o_fd[o]  = 1.0f;
    }
}

// Flat copy kernel for the coarse outputs: one thread per float4 of the
// gathered coarse rows; the tiny mask/score/dup vectors ride along.
__global__ __launch_bounds__(kThreads) void finer_coarse_kernel(
    const float* __restrict__ coarse_token_states,  // [B, NB, D]
    const float* __restrict__ coarse_token_mask,    // [B, NB]
    const int*   __restrict__ coarse_block_indices, // [B, NC]
    const float* __restrict__ coarse_block_scores,  // [B, NC]
    float* __restrict__ o_cs,                       // [B, NC, D]
    float* __restrict__ o_cm,                       // [B, NC]
    float* __restrict__ o_csc,                      // [B, NC]
    float* __restrict__ o_cd)                       // [B, NC]
{
    const int g   = blockIdx.x * kThreads + threadIdx.x;  // 0 .. B*NC*D4-1
    const int row = g >> 6;                               // kD4 == 64
    const int c4  = g & (kD4 - 1);
    const int b   = row / kNC;
    const int c   = row - b * kNC;

    const int j = coarse_block_indices[(size_t)b * kNC + c];
    const f4_t* __restrict__ src4 =
        (const f4_t*)(coarse_token_states + ((size_t)b * kNB + j) * kD);
    f4_t* __restrict__ dst4 =
        (f4_t*)(o_cs + ((size_t)b * kNC + c) * kD);
    // Gathered rows may repeat -> keep loads cacheable; output is write-once.
    __builtin_nontemporal_store(src4[c4], dst4 + c4);

    if (g < kB * kNC) {
        const int bb = g / kNC;
        const int cc = g - bb * kNC;
        const int jj = coarse_block_indices[(size_t)bb * kNC + cc];
        o_cm[g]  = coarse_token_mask[(size_t)bb * kNB + jj];
        o_csc[g] = coarse_block_scores[g];
        o_cd[g]  = (float)kBS;
    }
}

extern "C" void kernel_launch(void* const* d_in, const int* in_sizes, int n_in,
                              void* d_out, int out_size, void* d_ws, size_t ws_size,
                              hipStream_t stream) {
    (void)in_sizes; (void)n_in; (void)out_size; (void)d_ws; (void)ws_size;

    const float* fine_token_mask      = (const float*)d_in[0];
    const float* coarse_token_states  = (const float*)d_in[1];
    const float* coarse_token_mask    = (const float*)d_in[2];
    const int*   fine_block_indices   = (const int*)  d_in[3];
    const float* fine_block_scores    = (const float*)d_in[4];
    const int*   coarse_block_indices = (const int*)  d_in[5];
    const float* coarse_block_scores  = (const float*)d_in[6];
    const int*   cache_indice_table   = (const int*)  d_in[7];
    const float* cache_bank           = (const float*)d_in[8];

    float* out = (float*)d_out;
    float* o_fs  = out;                                        // B*NF*BS*D
    float* o_fm  = o_fs  + (size_t)kB * kNF * kBS * kD;        // B*NF*BS
    float* o_fsc = o_fm  + (size_t)kB * kNF * kBS;
    float* o_fd  = o_fsc + (size_t)kB * kNF * kBS;
    float* o_cs  = o_fd  + (size_t)kB * kNF * kBS;             // B*NC*D
    float* o_cm  = o_cs  + (size_t)kB * kNC * kD;
    float* o_csc = o_cm  + (size_t)kB * kNC;
    float* o_cd  = o_csc + (size_t)kB * kNC;

    finer_fine_kernel<<<kB * kNF, kThreads, 0, stream>>>(
        fine_token_mask, coarse_token_states,
        fine_block_indices, fine_block_scores,
        cache_indice_table, cache_bank,
        o_fs, o_fm, o_fsc, o_fd);

    const int coarse_threads_total = kB * kNC * kD4;           // 262144
    finer_coarse_kernel<<<coarse_threads_total / kThreads, kThreads, 0, stream>>>(
        coarse_token_states, coarse_token_mask,
        coarse_block_indices, coarse_block_scores,
        o_cs, o_cm, o_csc, o_cd);
}